// SparseGraphAttentionLayer_87668872446712
// MI455X (gfx1250) — compile-verified
//
#include <hip/hip_runtime.h>
#include <hip/hip_bf16.h>

typedef __attribute__((ext_vector_type(2))) float v2f;
typedef __attribute__((ext_vector_type(8))) float v8f;

#define N_NODES   50000
#define N_EDGES   800000
#define IN_F      256
#define HF        128   // NUM_HEADS * OUT_PER_HEAD
#define NH        8
#define FPH       16
#define SLOPE     0.2f

// ---------------------------------------------------------------------------
// Kernel 0: zero fill (d_out and denom are poisoned by the harness)
// ---------------------------------------------------------------------------
__global__ void zero_f32(float* __restrict__ p, int n) {
    int i = blockIdx.x * blockDim.x + threadIdx.x;
    if (i < n) p[i] = 0.0f;
}

// ---------------------------------------------------------------------------
// Kernel 1: h = x @ W^T + b  via V_WMMA_F32_16X16X4_F32  (fp32 exact path)
//
// Block = 256 threads = 8 waves. Block handles one 16-row M tile of x
// (staged in LDS, 16 KB), wave w computes the 16-col N tile [16w,16w+16).
// Per-wave: 64 chained WMMAs over K=256 in steps of 4.
//
// Operand layouts per CDNA5 ISA 7.12.2 (wave32):
//   A (16x4, M x K): lanes 0-15 & 16-31 both hold M=lane&15;
//                    VGPR0/1 = K = koff / koff+1, koff = (lane>=16) ? 2 : 0
//   B (4x16, K x N): lane gives N=lane&15; VGPR0/1 = K = koff / koff+1
//   C/D (16x16)    : VGPR i -> M = m0 + i + 8*(lane>=16), N = lane&15
// ---------------------------------------------------------------------------
__global__ __launch_bounds__(256) void gat_linear_wmma(
    const float* __restrict__ x, const float* __restrict__ W,
    const float* __restrict__ bias, float* __restrict__ h)
{
    __shared__ float xs[16 * IN_F];  // 16 KB of the 320 KB WGP LDS

    const int m0  = blockIdx.x * 16;        // 3125 * 16 == 50000 exactly
    const int tid = threadIdx.x;

    // cooperative 16x256 fp32 tile load, 128-bit per thread per trip
    {
        const float4* src = (const float4*)(x + (size_t)m0 * IN_F);
        float4*       dst = (float4*)xs;
        #pragma unroll
        for (int i = tid; i < 16 * IN_F / 4; i += 256) dst[i] = src[i];
    }
    __syncthreads();

    const int wave = tid >> 5;
    const int lane = tid & 31;
    const int half = lane >> 4;     // 0 = lanes 0-15, 1 = lanes 16-31
    const int l16  = lane & 15;
    const int n0   = wave * 16;
    const int koff = half * 2;

    const float* arow = xs + l16 * IN_F;                 // A: row m0+l16 of x
    const float* brow = W + (size_t)(n0 + l16) * IN_F;   // B col n = W row n

    v8f acc = {};
    #pragma unroll 8
    for (int k0 = 0; k0 < IN_F; k0 += 4) {
        v2f a = *(const v2f*)(arow + k0 + koff);   // K = k0+koff, k0+koff+1
        v2f b = *(const v2f*)(brow + k0 + koff);
        acc = __builtin_amdgcn_wmma_f32_16x16x4_f32(
            /*neg_a=*/false, a, /*neg_b=*/false, b,
            /*c_mod=*/(short)0, acc, /*reuse_a=*/false, /*reuse_b=*/false);
    }

    const int   n  = n0 + l16;
    const float bv = bias[n];
    #pragma unroll
    for (int i = 0; i < 8; ++i) {
        int m = m0 + i + half * 8;
        h[(size_t)m * HF + n] = acc[i] + bv;
    }
}

// ---------------------------------------------------------------------------
// Kernel 2: per-(node, head) attention logits
//   s_row[n,h] = sum_f h[n,h,f] * attn_w[h,f,0]
//   s_col[n,h] = sum_f h[n,h,f] * attn_w[h,f,1]
// note (idx/8)*128 + (idx%8)*16 == idx*16
// ---------------------------------------------------------------------------
__global__ void gat_attn_logits(
    const float* __restrict__ h, const float* __restrict__ attn_w,
    float* __restrict__ s_row, float* __restrict__ s_col)
{
    int idx = blockIdx.x * blockDim.x + threadIdx.x;   // n*8 + head
    if (idx >= N_NODES * NH) return;
    int head = idx & 7;
    const float* hp = h + (size_t)idx * FPH;
    const float* aw = attn_w + head * (FPH * 2);       // [f][k], k in {0,1}
    float sr = 0.0f, sc = 0.0f;
    #pragma unroll
    for (int f = 0; f < FPH; ++f) {
        float hv = hp[f];
        sr += hv * aw[2 * f + 0];
        sc += hv * aw[2 * f + 1];
    }
    s_row[idx] = sr;
    s_col[idx] = sc;
}

// ---------------------------------------------------------------------------
// Kernel 3: per-edge exp(leaky_relu(logit)) + per-(row,head) denominator
// One thread per edge; 8 heads via float4 pairs; atomic f32 adds land in L2.
// ---------------------------------------------------------------------------
__global__ void gat_edge_softmax_num(
    const int* __restrict__ ei, const float* __restrict__ s_row,
    const float* __restrict__ s_col, float* __restrict__ wbuf,
    float* __restrict__ denom)
{
    int e = blockIdx.x * blockDim.x + threadIdx.x;
    if (e >= N_EDGES) return;
    int r = ei[e];              // destination (row)
    int c = ei[N_EDGES + e];    // source (col)

    const float4* srp = (const float4*)(s_row + (size_t)r * NH);
    const float4* scp = (const float4*)(s_col + (size_t)c * NH);
    float4 a0 = srp[0], a1 = srp[1];
    float4 b0 = scp[0], b1 = scp[1];

    float v[NH] = { a0.x + b0.x, a0.y + b0.y, a0.z + b0.z, a0.w + b0.w,
                    a1.x + b1.x, a1.y + b1.y, a1.z + b1.z, a1.w + b1.w };
    #pragma unroll
    for (int k = 0; k < NH; ++k) {
        float t = v[k];
        t = (t > 0.0f) ? t : SLOPE * t;    // leaky_relu, slope 0.2
        float w = __expf(t);               // softmax ratio cancels ulp error
        wbuf[(size_t)e * NH + k] = w;
        atomicAdd(&denom[(size_t)r * NH + k], w);
    }
}

// ---------------------------------------------------------------------------
// Kernel 4: out[row, :] += (w/denom[row]) * h[col, :]
// 128 threads per edge (2 edges per 256-thread block): the h[col] gather is a
// coalesced 512B burst and the atomic scatter to out[row] is coalesced too.
// out (25.6 MB) stays resident in the 192 MB L2, so atomics resolve in L2.
// ---------------------------------------------------------------------------
__global__ __launch_bounds__(256) void gat_aggregate(
    const int* __restrict__ ei, const float* __restrict__ h,
    const float* __restrict__ wbuf, const float* __restrict__ denom,
    float* __restrict__ out)
{
    int t    = threadIdx.x & 127;                  // feature 0..127
    int e    = blockIdx.x * 2 + (threadIdx.x >> 7);
    int r    = ei[e];
    int c    = ei[N_EDGES + e];
    int head = t >> 4;
    float alpha = wbuf[(size_t)e * NH + head] / denom[(size_t)r * NH + head];
    float msg   = alpha * h[(size_t)c * HF + t];
    atomicAdd(&out[(size_t)r * HF + t], msg);
}

// ---------------------------------------------------------------------------
// Launch
// inputs: 0=x [N,256] f32, 1=W [128,256] f32, 2=b [128] f32,
//         3=attn_w [8,16,2] f32, 4=edge_index [2,E] i32
// ---------------------------------------------------------------------------
extern "C" void kernel_launch(void* const* d_in, const int* in_sizes, int n_in,
                              void* d_out, int out_size, void* d_ws, size_t ws_size,
                              hipStream_t stream) {
    const float* x      = (const float*)d_in[0];
    const float* W      = (const float*)d_in[1];
    const float* bias   = (const float*)d_in[2];
    const float* attn_w = (const float*)d_in[3];
    const int*   ei     = (const int*)d_in[4];
    float*       out    = (float*)d_out;

    // workspace layout (floats)
    float* h     = (float*)d_ws;                   // N*128  = 6,400,000
    float* s_row = h     + (size_t)N_NODES * HF;   // N*8    =   400,000
    float* s_col = s_row + (size_t)N_NODES * NH;   // N*8
    float* wbuf  = s_col + (size_t)N_NODES * NH;   // E*8    = 6,400,000
    float* denom = wbuf  + (size_t)N_EDGES * NH;   // N*8

    // zero accumulators (harness poisons d_out / d_ws)
    zero_f32<<<(N_NODES * HF + 255) / 256, 256, 0, stream>>>(out, N_NODES * HF);
    zero_f32<<<(N_NODES * NH + 255) / 256, 256, 0, stream>>>(denom, N_NODES * NH);

    // 1) dense projection via fp32 WMMA
    gat_linear_wmma<<<N_NODES / 16, 256, 0, stream>>>(x, W, bias, h);

    // 2) per-(node,head) logits
    gat_attn_logits<<<(N_NODES * NH + 255) / 256, 256, 0, stream>>>(
        h, attn_w, s_row, s_col);

    // 3) edge numerators + segment-sum denominators
    gat_edge_softmax_num<<<(N_EDGES + 255) / 256, 256, 0, stream>>>(
        ei, s_row, s_col, wbuf, denom);

    // 4) normalized aggregation (2 edges per block)
    gat_aggregate<<<N_EDGES / 2, 256, 0, stream>>>(ei, h, wbuf, denom, out);
}